// VQVAEModel_30288109371653
// MI455X (gfx1250) — compile-verified
//
#include <hip/hip_runtime.h>

typedef __attribute__((ext_vector_type(2))) float v2f;
typedef __attribute__((ext_vector_type(8))) float v8f;

#define N_ROWS 16384   // 16*32*32
#define DIM    32
#define K_CODES 8192
#define OUT_ELEMS (N_ROWS * DIM)
#define K_SPLITS 4
#define K_PER_SPLIT (K_CODES / K_SPLITS)

// order-preserving float <-> uint key (monotone: a<b  <=>  f2key(a)<f2key(b))
__device__ __forceinline__ unsigned f2key(float f) {
    unsigned b = __float_as_uint(f);
    return (b & 0x80000000u) ? ~b : (b | 0x80000000u);
}
__device__ __forceinline__ float key2f(unsigned k) {
    unsigned b = (k & 0x80000000u) ? (k & 0x7fffffffu) : ~k;
    return __uint_as_float(b);
}

__device__ __forceinline__ unsigned long long shfl_xor_u64(unsigned long long v, int mask) {
    unsigned lo = (unsigned)v, hi = (unsigned)(v >> 32);
    lo = __shfl_xor(lo, mask, 32);
    hi = __shfl_xor(hi, mask, 32);
    return ((unsigned long long)hi << 32) | lo;
}

// ---------------------------------------------------------------------------
// Kernel 1: norms + workspace init
// ---------------------------------------------------------------------------
__global__ void vq_init(const float* __restrict__ x, const float* __restrict__ cb,
                        float* __restrict__ cc, float* __restrict__ xx,
                        unsigned long long* __restrict__ rowkey,
                        unsigned* __restrict__ dmin, unsigned* __restrict__ used,
                        float* __restrict__ accum) {
    int i = blockIdx.x * blockDim.x + threadIdx.x;
    if (i < K_CODES) {
        const float4* r = (const float4*)(cb + i * DIM);
        float s = 0.f;
#pragma unroll
        for (int t = 0; t < 8; ++t) {
            float4 v = r[t];
            s += v.x * v.x + v.y * v.y + v.z * v.z + v.w * v.w;
        }
        cc[i] = s;
        dmin[i] = 0xFFFFFFFFu;  // +inf key
        used[i] = 0u;
    }
    if (i < N_ROWS) {
        const float4* r = (const float4*)(x + i * DIM);
        float s = 0.f;
#pragma unroll
        for (int t = 0; t < 8; ++t) {
            float4 v = r[t];
            s += v.x * v.x + v.y * v.y + v.z * v.z + v.w * v.w;
        }
        xx[i] = s;
        rowkey[i] = 0xFFFFFFFFFFFFFFFFull;
    }
    if (i == 0) { accum[0] = 0.f; accum[1] = 0.f; }
}

// ---------------------------------------------------------------------------
// Kernel 2: WMMA distance GEMM + fused row-argmin / column-min.
// Grid = (N/32 row-slabs) x (4 K-splits), 1 wave per block, 32 rows per wave.
// A is pre-scaled by -2 and C seeded with ||c||^2 so WMMA emits the distance
// (minus the row-constant ||x||^2) directly.
// ---------------------------------------------------------------------------
__global__ void __launch_bounds__(32)
vq_argmin(const float* __restrict__ x, const float* __restrict__ cb,
          const float* __restrict__ cc, const float* __restrict__ xx,
          unsigned long long* __restrict__ rowkey, unsigned* __restrict__ dmin) {
    const int lane  = threadIdx.x;
    const int half  = lane >> 4;   // 0 or 1
    const int col   = lane & 15;
    const int tm    = (blockIdx.x >> 2) * 32;          // row-slab base
    const int kbase = (blockIdx.x & 3) * K_PER_SPLIT;  // K-split base

    // Preload A (inputs, scaled by -2) for both 16-row slabs; f32 16x16x4 A layout:
    // A[lane][vj] = x[row = slab*16 + lane%16][k = kk + 2*(lane/16) + vj]
    v2f a[2][8];
    float xr[2][8];
#pragma unroll
    for (int s = 0; s < 2; ++s) {
#pragma unroll
        for (int t = 0; t < 8; ++t) {
            v2f v = *(const v2f*)(x + (size_t)(tm + s * 16 + col) * DIM + 4 * t + 2 * half);
            a[s][t] = -2.f * v;
            xr[s][t] = xx[tm + s * 16 + 8 * half + t];  // row of C vgpr t
        }
    }

    float rm[2][8];
    int   ri[2][8];
#pragma unroll
    for (int s = 0; s < 2; ++s)
#pragma unroll
        for (int j = 0; j < 8; ++j) { rm[s][j] = __builtin_huge_valf(); ri[s][j] = 0; }

    for (int kt = kbase; kt < kbase + K_PER_SPLIT; kt += 16) {
        // B tile: B[lane][vj] = cb[code = kt + lane%16][k = kk + 2*(lane/16) + vj]
        v2f b[8];
#pragma unroll
        for (int t = 0; t < 8; ++t)
            b[t] = *(const v2f*)(cb + (size_t)(kt + col) * DIM + 4 * t + 2 * half);

        const int   code = kt + col;
        const float ccv  = cc[code];
        // seed C with ||c||^2 splat (per-column constant)
        v8f acc0 = {ccv, ccv, ccv, ccv, ccv, ccv, ccv, ccv};
        v8f acc1 = acc0;
#pragma unroll
        for (int t = 0; t < 8; ++t) {
            acc0 = __builtin_amdgcn_wmma_f32_16x16x4_f32(false, a[0][t], false, b[t],
                                                         (short)0, acc0, false, false);
            acc1 = __builtin_amdgcn_wmma_f32_16x16x4_f32(false, a[1][t], false, b[t],
                                                         (short)0, acc1, false, false);
        }

        float colm = __builtin_huge_valf();
#pragma unroll
        for (int j = 0; j < 8; ++j) {
            float d0 = acc0[j];   // ||c||^2 - 2 x.c
            float d1 = acc1[j];
            if (d0 < rm[0][j]) { rm[0][j] = d0; ri[0][j] = code; }
            if (d1 < rm[1][j]) { rm[1][j] = d1; ri[1][j] = code; }
            colm = fminf(colm, d0 + xr[0][j]);   // full distance for entropy term
            colm = fminf(colm, d1 + xr[1][j]);
        }
        // Per-code column min: fire-and-forget non-returning atomic from every
        // lane (lane pairs col/col+16 cover all 32 rows of this slab).
        // No shuffle / EXEC predication -> keeps ds ops and waits out of the loop;
        // STOREcnt is only drained at s_endpgm.
        atomicMin(&dmin[code], f2key(colm));
    }

    // Row argmin: pack (dist,code) into u64 key, reduce across the 16 lanes of
    // each half, then merge across K-splits with a 64-bit atomicMin.
#pragma unroll
    for (int s = 0; s < 2; ++s) {
#pragma unroll
        for (int j = 0; j < 8; ++j) {
            unsigned long long key =
                ((unsigned long long)f2key(rm[s][j]) << 32) | (unsigned)ri[s][j];
#pragma unroll
            for (int off = 1; off < 16; off <<= 1) {
                unsigned long long ok = shfl_xor_u64(key, off);
                key = (ok < key) ? ok : key;
            }
            if (col == 0) {
                int row = tm + s * 16 + 8 * half + j;
                atomicMin(&rowkey[row], key);
            }
        }
    }
}

// ---------------------------------------------------------------------------
// Kernel 3: extract tokens, gather emb -> out, mark used codes,
//           accumulate SSE for embedding+commitment loss
// ---------------------------------------------------------------------------
__global__ void vq_emb(const float* __restrict__ x, const float* __restrict__ cb,
                       const unsigned long long* __restrict__ rowkey,
                       unsigned* __restrict__ used,
                       float* __restrict__ out, float* __restrict__ accum) {
    int n = blockIdx.x * blockDim.x + threadIdx.x;
    float s = 0.f;
    if (n < N_ROWS) {
        int t = (int)(rowkey[n] & 0xFFFFFFFFull);
        used[t] = 1u;
        const float4* cr = (const float4*)(cb + (size_t)t * DIM);
        const float4* xrow = (const float4*)(x + (size_t)n * DIM);
        float4* orow = (float4*)(out + (size_t)n * DIM);
#pragma unroll
        for (int i = 0; i < 8; ++i) {
            float4 c = cr[i];
            float4 v = xrow[i];
            orow[i] = c;  // straight-through forward value == emb
            float dx = c.x - v.x, dy = c.y - v.y, dz = c.z - v.z, dw = c.w - v.w;
            s += dx * dx + dy * dy + dz * dz + dw * dw;
        }
    }
#pragma unroll
    for (int off = 1; off < 32; off <<= 1) s += __shfl_xor(s, off, 32);
    if ((threadIdx.x & 31) == 0) atomicAdd(&accum[0], s);
}

// ---------------------------------------------------------------------------
// Kernel 4: entropy term: sum of min-dist over unused codes
// ---------------------------------------------------------------------------
__global__ void vq_entropy(const unsigned* __restrict__ dmin,
                           const unsigned* __restrict__ used,
                           float* __restrict__ accum) {
    int k = blockIdx.x * blockDim.x + threadIdx.x;
    float e = 0.f;
    if (k < K_CODES && used[k] == 0u) e = key2f(dmin[k]);
#pragma unroll
    for (int off = 1; off < 32; off <<= 1) e += __shfl_xor(e, off, 32);
    if ((threadIdx.x & 31) == 0) atomicAdd(&accum[1], e);
}

// ---------------------------------------------------------------------------
// Kernel 5: finalize scalar loss
// ---------------------------------------------------------------------------
__global__ void vq_finalize(const float* __restrict__ accum, float* __restrict__ out) {
    // (embedding 1.0 + commitment 0.25) * mean + 0.1 * mean-over-K entropy
    out[OUT_ELEMS] = 1.25f * accum[0] * (1.0f / (float)OUT_ELEMS)
                   + 0.1f  * accum[1] * (1.0f / (float)K_CODES);
}

extern "C" void kernel_launch(void* const* d_in, const int* in_sizes, int n_in,
                              void* d_out, int out_size, void* d_ws, size_t ws_size,
                              hipStream_t stream) {
    const float* x  = (const float*)d_in[0];  // [16,32,32,32] -> [16384,32]
    const float* cb = (const float*)d_in[1];  // [8192,32]
    float* out = (float*)d_out;               // [524288] emb + [1] loss

    // workspace layout (rowkey first for 8-byte alignment)
    char* w = (char*)d_ws;
    unsigned long long* rowkey = (unsigned long long*)w;             // N_ROWS u64
    unsigned* dmin = (unsigned*)(w + sizeof(unsigned long long) * N_ROWS); // K u32
    unsigned* used = dmin + K_CODES;                                  // K u32
    float*    cc   = (float*)(used + K_CODES);                        // K f32
    float*    xx   = cc + K_CODES;                                    // N f32
    float*    accum = xx + N_ROWS;                                    // 2 f32

    (void)in_sizes; (void)n_in; (void)out_size; (void)ws_size;

    vq_init<<<N_ROWS / 256, 256, 0, stream>>>(x, cb, cc, xx, rowkey, dmin, used, accum);
    vq_argmin<<<(N_ROWS / 32) * K_SPLITS, 32, 0, stream>>>(x, cb, cc, xx, rowkey, dmin);
    vq_emb<<<N_ROWS / 256, 256, 0, stream>>>(x, cb, rowkey, used, out, accum);
    vq_entropy<<<K_CODES / 256, 256, 0, stream>>>(dmin, used, accum);
    vq_finalize<<<1, 1, 0, stream>>>(accum, out);
}